// Encoder_60361470378157
// MI455X (gfx1250) — compile-verified
//
#include <hip/hip_runtime.h>
#include <hip/hip_bf16.h>

typedef float v2f __attribute__((ext_vector_type(2)));
typedef float v8f __attribute__((ext_vector_type(8)));

#define N_NODES 50000
#define N_EDGES 800000
#define CH      256          // dout for both layers
#define BN_EPS  1e-5f

// ---------------------------------------------------------------------------
// Degree / normalization
// ---------------------------------------------------------------------------
__global__ void deg_init_kernel(float* deg, int n) {
    int i = blockIdx.x * blockDim.x + threadIdx.x;
    if (i < n) deg[i] = 1.0f;   // self-loop weight
}

__global__ void deg_count_kernel(const long long* __restrict__ ei, float* deg, int E) {
    int e = blockIdx.x * blockDim.x + threadIdx.x;
    if (e < E) atomicAdd(&deg[(int)ei[(size_t)E + e]], 1.0f);
}

__global__ void deg_rsqrt_kernel(float* deg, int n) {
    int i = blockIdx.x * blockDim.x + threadIdx.x;
    if (i < n) deg[i] = rsqrtf(deg[i]);   // deg >= 1 always
}

// ---------------------------------------------------------------------------
// H = X @ W  (X: [N,K] f32, W: [K,256] f32, H: [N,256] f32)
// block = 128 threads = 4 waves; each block computes a 16x256 strip.
// A-tile (16xK) staged in LDS; each wave owns four 16x16 N-tiles.
// Uses V_WMMA_F32_16X16X4_F32.
// ---------------------------------------------------------------------------
__global__ void __launch_bounds__(128)
gemm_wmma_f32_kernel(const float* __restrict__ X, const float* __restrict__ W,
                     float* __restrict__ H, int K) {
    __shared__ float sA[16 * 256];

    const int tid  = threadIdx.x;
    const int wave = tid >> 5;        // 0..3
    const int lane = tid & 31;
    const int r    = lane & 15;       // row within tile / col within tile
    const int hi   = lane >> 4;       // half-wave select (K pair / M half)
    const int row_base = blockIdx.x * 16;

    // Cooperative, coalesced float4 load of the 16xK A tile into LDS.
    const int nvec = (16 * K) >> 2;
    const float4* Xv  = (const float4*)(X + (size_t)row_base * K);
    float4*       sAv = (float4*)sA;
    for (int i = tid; i < nvec; i += 128) sAv[i] = Xv[i];
    __syncthreads();

    const int col_base = wave * 64;

    v8f acc0 = {}, acc1 = {}, acc2 = {}, acc3 = {};

    for (int k0 = 0; k0 < K; k0 += 4) {
        // A fragment: VGPR0 holds K = k0+2*hi, VGPR1 holds K = k0+2*hi+1
        v2f a;
        a.x = sA[r * K + k0 + 2 * hi];
        a.y = sA[r * K + k0 + 2 * hi + 1];

        // B fragments: row K striped across lanes within a VGPR
        const float* Wr0 = W + (size_t)(k0 + 2 * hi) * CH + col_base + r;
        const float* Wr1 = Wr0 + CH;
        v2f b0; b0.x = Wr0[0];  b0.y = Wr1[0];
        v2f b1; b1.x = Wr0[16]; b1.y = Wr1[16];
        v2f b2; b2.x = Wr0[32]; b2.y = Wr1[32];
        v2f b3; b3.x = Wr0[48]; b3.y = Wr1[48];

        acc0 = __builtin_amdgcn_wmma_f32_16x16x4_f32(false, a, false, b0, (short)0, acc0, false, false);
        acc1 = __builtin_amdgcn_wmma_f32_16x16x4_f32(false, a, false, b1, (short)0, acc1, false, false);
        acc2 = __builtin_amdgcn_wmma_f32_16x16x4_f32(false, a, false, b2, (short)0, acc2, false, false);
        acc3 = __builtin_amdgcn_wmma_f32_16x16x4_f32(false, a, false, b3, (short)0, acc3, false, false);
    }

    // D layout: VGPR rr -> row rr + 8*hi, col = lane&15
    float* Hb = H + (size_t)row_base * CH + col_base + r;
    #pragma unroll
    for (int rr = 0; rr < 8; ++rr) {
        float* dst = Hb + (size_t)(rr + 8 * hi) * CH;
        dst[0]  = acc0[rr];
        dst[16] = acc1[rr];
        dst[32] = acc2[rr];
        dst[48] = acc3[rr];
    }
}

// ---------------------------------------------------------------------------
// out[dst] += h[src] * (dinv[src]*dinv[dst])  over E edges + N self-loops.
// 64 threads per edge, float4 per thread, scattered via global f32 atomics.
// ---------------------------------------------------------------------------
__global__ void aggregate_kernel(const float* __restrict__ h,
                                 const long long* __restrict__ ei,
                                 const float* __restrict__ dinv,
                                 float* __restrict__ out,
                                 int E, int n) {
    long long gid  = (long long)blockIdx.x * blockDim.x + threadIdx.x;
    long long edge = gid >> 6;
    int       lane = (int)(gid & 63);
    if (edge >= (long long)(E + n)) return;

    int src, dst;
    if (edge < E) {
        src = (int)ei[edge];
        dst = (int)ei[(size_t)E + edge];
    } else {
        src = dst = (int)(edge - E);
    }
    float norm = dinv[src] * dinv[dst];

    float4 hv = ((const float4*)(h + (size_t)src * CH))[lane];
    float* o  = out + (size_t)dst * CH + lane * 4;
    atomicAdd(o + 0, hv.x * norm);
    atomicAdd(o + 1, hv.y * norm);
    atomicAdd(o + 2, hv.z * norm);
    atomicAdd(o + 3, hv.w * norm);
}

// ---------------------------------------------------------------------------
// BatchNorm statistics over nodes (bias folded in), then fused apply + PReLU.
// ---------------------------------------------------------------------------
__global__ void bn_stats_kernel(const float* __restrict__ v, const float* __restrict__ bias,
                                float* gsum, float* gsumsq, int n) {
    int ch = threadIdx.x;                  // 256 channels
    float b = bias[ch];
    float s = 0.0f, s2 = 0.0f;
    for (int row = blockIdx.x; row < n; row += gridDim.x) {
        float x = v[(size_t)row * CH + ch] + b;
        s  += x;
        s2 += x * x;
    }
    atomicAdd(&gsum[ch], s);
    atomicAdd(&gsumsq[ch], s2);
}

__global__ void bn_finalize_kernel(const float* gsum, const float* gsumsq,
                                   const float* gamma, const float* beta,
                                   float* scale, float* shift, float n) {
    int c = threadIdx.x;
    float mean = gsum[c] / n;
    float var  = gsumsq[c] / n - mean * mean;
    float inv  = rsqrtf(var + BN_EPS);
    float sc   = gamma[c] * inv;
    scale[c] = sc;
    shift[c] = beta[c] - mean * sc;
}

__global__ void bn_prelu_apply_kernel(const float* __restrict__ v, const float* __restrict__ bias,
                                      const float* __restrict__ scale, const float* __restrict__ shift,
                                      const float* __restrict__ a, float* __restrict__ out,
                                      long long total) {
    long long i = (long long)blockIdx.x * blockDim.x + threadIdx.x;
    if (i >= total) return;
    int c = (int)(i & (CH - 1));
    float y = (v[i] + bias[c]) * scale[c] + shift[c];
    float alpha = a[0];
    out[i] = y > 0.0f ? y : alpha * y;
}

// ---------------------------------------------------------------------------
// Launch
// ---------------------------------------------------------------------------
extern "C" void kernel_launch(void* const* d_in, const int* in_sizes, int n_in,
                              void* d_out, int out_size, void* d_ws, size_t ws_size,
                              hipStream_t stream) {
    (void)in_sizes; (void)n_in; (void)out_size; (void)ws_size;

    const float*     x  = (const float*)d_in[0];
    const long long* ei = (const long long*)d_in[1];   // int64 edge_index [2,E]
    const float* W0 = (const float*)d_in[2];
    const float* b0 = (const float*)d_in[3];
    const float* g0 = (const float*)d_in[4];
    const float* be0 = (const float*)d_in[5];
    const float* a0 = (const float*)d_in[6];
    const float* W1 = (const float*)d_in[7];
    const float* b1 = (const float*)d_in[8];
    const float* g1 = (const float*)d_in[9];
    const float* be1 = (const float*)d_in[10];
    const float* a1 = (const float*)d_in[11];
    float* out = (float*)d_out;

    const int N = N_NODES, E = N_EDGES;
    const long long NC = (long long)N * CH;

    // Workspace carve-up (floats)
    float* ws     = (float*)d_ws;
    float* dinv   = ws;                       // N
    float* gsum   = ws + N;                   // 256
    float* gsumsq = gsum + CH;                // 256
    float* scale  = gsumsq + CH;              // 256
    float* shift  = scale + CH;               // 256
    float* bufA   = shift + CH;               // N*256
    float* bufB   = bufA + NC;                // N*256

    const int T = 256;
    dim3 blkElt(T);

    // --- symmetric normalization (shared across layers) -------------------
    deg_init_kernel <<<(N + T - 1) / T, blkElt, 0, stream>>>(dinv, N);
    deg_count_kernel<<<(E + T - 1) / T, blkElt, 0, stream>>>(ei, dinv, E);
    deg_rsqrt_kernel<<<(N + T - 1) / T, blkElt, 0, stream>>>(dinv, N);

    const int gemmBlocks = N / 16;                              // 3125 (N divisible by 16)
    const long long aggThreads = (long long)(E + N) * 64;
    const int aggBlocks  = (int)((aggThreads + T - 1) / T);
    const int applyBlocks = (int)((NC + T - 1) / T);

    // ============================ Layer 1 =================================
    // h0 = x @ W0  -> bufB
    gemm_wmma_f32_kernel<<<gemmBlocks, 128, 0, stream>>>(x, W0, bufB, 128);
    // agg0 -> bufA
    hipMemsetAsync(bufA, 0, (size_t)NC * sizeof(float), stream);
    aggregate_kernel<<<aggBlocks, blkElt, 0, stream>>>(bufB, ei, dinv, bufA, E, N);
    // BN + PReLU -> bufB (activations)
    hipMemsetAsync(gsum, 0, 2 * CH * sizeof(float), stream);
    bn_stats_kernel   <<<256, CH, 0, stream>>>(bufA, b0, gsum, gsumsq, N);
    bn_finalize_kernel<<<1,   CH, 0, stream>>>(gsum, gsumsq, g0, be0, scale, shift, (float)N);
    bn_prelu_apply_kernel<<<applyBlocks, blkElt, 0, stream>>>(bufA, b0, scale, shift, a0, bufB, NC);

    // ============================ Layer 2 =================================
    // h1 = act @ W1 -> bufA
    gemm_wmma_f32_kernel<<<gemmBlocks, 128, 0, stream>>>(bufB, W1, bufA, 256);
    // agg1 -> bufB
    hipMemsetAsync(bufB, 0, (size_t)NC * sizeof(float), stream);
    aggregate_kernel<<<aggBlocks, blkElt, 0, stream>>>(bufA, ei, dinv, bufB, E, N);
    // BN + PReLU -> d_out
    hipMemsetAsync(gsum, 0, 2 * CH * sizeof(float), stream);
    bn_stats_kernel   <<<256, CH, 0, stream>>>(bufB, b1, gsum, gsumsq, N);
    bn_finalize_kernel<<<1,   CH, 0, stream>>>(gsum, gsumsq, g1, be1, scale, shift, (float)N);
    bn_prelu_apply_kernel<<<applyBlocks, blkElt, 0, stream>>>(bufB, b1, scale, shift, a1, out, NC);
}